// GeometryLocationAttentionHead_47579647705900
// MI455X (gfx1250) — compile-verified
//
#include <hip/hip_runtime.h>
#include <math.h>

// ---------------------------------------------------------------------------
// GeometryLocationAttentionHead for MI455X (gfx1250, wave32, WMMA)
//
// Factorization: raw[e] = A[idx0] + B[idx1] + 6 pos-diff/frame silu terms.
// A/B are per-node reductions over WMMA-computed projections.
// ---------------------------------------------------------------------------

typedef float v2f __attribute__((ext_vector_type(2)));
typedef float v8f __attribute__((ext_vector_type(8)));

__device__ __forceinline__ float silu_f(float x) { return x / (1.0f + expf(-x)); }

__device__ __forceinline__ v8f wmma4(v2f a, v2f b, v8f c) {
  // D = A(16x4 f32) * B(4x16 f32) + C(16x16 f32)
  return __builtin_amdgcn_wmma_f32_16x16x4_f32(false, a, false, b, (short)0, c,
                                               false, false);
}

// monotone float <-> uint bijection so unsigned atomicMax == float max
__device__ __forceinline__ unsigned f32_ord(float f) {
  unsigned u = __float_as_uint(f);
  return (u & 0x80000000u) ? ~u : (u | 0x80000000u);
}
__device__ __forceinline__ float ord_f32(unsigned u) {
  unsigned b = (u & 0x80000000u) ? (u ^ 0x80000000u) : ~u;
  return __uint_as_float(b);
}

#define WAVES_PER_BLK 4
#define NODE_SLOT 160  // per-node LDS floats: fs32 | ts32 | fv48 | tv48

// ---------------------------------------------------------------------------
// Kernel 1: per-node WMMA projections + silu-weighted reduction -> A[i], B[i]
// One wave = one 16-node tile.
// ---------------------------------------------------------------------------
__global__ __launch_bounds__(32 * WAVES_PER_BLK)
void node_precompute(const float* __restrict__ from_s, const float* __restrict__ from_v,
                     const float* __restrict__ to_s,   const float* __restrict__ to_v,
                     const float* __restrict__ from_frame, const float* __restrict__ to_frame,
                     const float* __restrict__ Wfs, const float* __restrict__ Wts,
                     const float* __restrict__ Wfv, const float* __restrict__ Wtv,
                     const float* __restrict__ Wattn,
                     float* __restrict__ Anode, float* __restrict__ Bnode,
                     int Nn, int NT)
{
  __shared__ float lds[WAVES_PER_BLK][16 * NODE_SLOT];  // 40 KB

  const int lane = threadIdx.x & 31;
  const int w    = threadIdx.x >> 5;
  const int tile = blockIdx.x * WAVES_PER_BLK + w;
  const bool live = (tile < NT);  // wave-uniform: EXEC stays all-ones

  if (live) {
    const int mrow = lane & 15;          // node-in-tile (A fragment M)
    const int koff = (lane >> 4) * 2;    // K sub-offset 0 / 2
    int row = tile * 16 + mrow;
    if (row >= Nn) row = Nn - 1;         // clamp tail loads (writes guarded later)
    const int col = lane & 15;           // output column (h)

    // ---- scalar GEMMs: [16,64] x [64,32] for from_s/Wfs and to_s/Wts ----
    v8f fs0 = {}, fs1 = {}, ts0 = {}, ts1 = {};
#pragma unroll
    for (int k0 = 0; k0 < 64; k0 += 4) {
      const int k = k0 + koff;
      v2f af  = *(const v2f*)(from_s + row * 64 + k);
      v2f at  = *(const v2f*)(to_s   + row * 64 + k);
      v2f bf0 = *(const v2f*)(Wfs + (col)      * 64 + k);
      v2f bf1 = *(const v2f*)(Wfs + (col + 16) * 64 + k);
      v2f bt0 = *(const v2f*)(Wts + (col)      * 64 + k);
      v2f bt1 = *(const v2f*)(Wts + (col + 16) * 64 + k);
      fs0 = wmma4(af, bf0, fs0);
      fs1 = wmma4(af, bf1, fs1);
      ts0 = wmma4(at, bt0, ts0);
      ts1 = wmma4(at, bt1, ts1);
    }

    // ---- vector GEMMs per xyz channel: [16,16] x [16,16] ----
    v8f fv[3], tv[3];
#pragma unroll
    for (int c = 0; c < 3; ++c) { fv[c] = {}; tv[c] = {}; }
#pragma unroll
    for (int k0 = 0; k0 < 16; k0 += 4) {
      const int k = k0 + koff;
#pragma unroll
      for (int c = 0; c < 3; ++c) {
        v2f afv, atv;
        afv.x = from_v[row * 48 + (k    ) * 3 + c];
        afv.y = from_v[row * 48 + (k + 1) * 3 + c];
        atv.x = to_v  [row * 48 + (k    ) * 3 + c];
        atv.y = to_v  [row * 48 + (k + 1) * 3 + c];
        v2f bfv = *(const v2f*)(Wfv + col * 16 + k);
        v2f btv = *(const v2f*)(Wtv + col * 16 + k);
        fv[c] = wmma4(afv, bfv, fv[c]);
        tv[c] = wmma4(atv, btv, tv[c]);
      }
    }

    // ---- spill fragments to LDS in [node][slot] layout ----
    float* L = &lds[w][0];
    const int mBase = (lane >> 4) * 8;  // C layout: lanes16-31 hold M=r+8
    const int h = lane & 15;
#pragma unroll
    for (int r = 0; r < 8; ++r) {
      const int m = r + mBase;
      float* Lm = L + m * NODE_SLOT;
      Lm[h]            = fs0[r];
      Lm[16 + h]       = fs1[r];
      Lm[32 + h]       = ts0[r];
      Lm[48 + h]       = ts1[r];
#pragma unroll
      for (int c = 0; c < 3; ++c) {
        Lm[64  + c * 16 + h] = fv[c][r];
        Lm[112 + c * 16 + h] = tv[c][r];
      }
    }
  }

  __syncthreads();

  if (live) {
    // 2 lanes per node: p selects which half of h-space each lane reduces
    const int p = lane & 1;
    const int m = lane >> 1;
    const int node = tile * 16 + m;
    const int nodec = (node < Nn) ? node : (Nn - 1);
    const float* Lm = &lds[w][m * NODE_SLOT];

    float accA = 0.0f, accB = 0.0f;

    // fs / ts contributions (Wattn[0:32], Wattn[32:64])
#pragma unroll
    for (int hh = 0; hh < 16; ++hh) {
      const int h = p * 16 + hh;
      accA += silu_f(Lm[h])      * Wattn[h];
      accB += silu_f(Lm[32 + h]) * Wattn[32 + h];
    }

    // per-node frames (row-major [c][d])
    float F[9], T[9];
#pragma unroll
    for (int q = 0; q < 9; ++q) { F[q] = from_frame[nodec * 9 + q]; T[q] = to_frame[nodec * 9 + q]; }

    // fv_scal / tv_scal: sum_c fv[c,h]*frame[c,d] -> silu -> Wattn[64+h*3+d] / [115+...]
#pragma unroll
    for (int hh = 0; hh < 8; ++hh) {
      const int h = p * 8 + hh;
      const float f0 = Lm[64 + h], f1 = Lm[80 + h], f2 = Lm[96 + h];
      const float t0 = Lm[112 + h], t1 = Lm[128 + h], t2 = Lm[144 + h];
#pragma unroll
      for (int d = 0; d < 3; ++d) {
        const float sf = f0 * F[d] + f1 * F[3 + d] + f2 * F[6 + d];
        const float st = t0 * T[d] + t1 * T[3 + d] + t2 * T[6 + d];
        accA += silu_f(sf) * Wattn[64  + h * 3 + d];
        accB += silu_f(st) * Wattn[115 + h * 3 + d];
      }
    }

    accA += __shfl_xor(accA, 1, 32);
    accB += __shfl_xor(accB, 1, 32);
    if (p == 0 && node < Nn) { Anode[node] = accA; Bnode[node] = accB; }
  }
}

// ---------------------------------------------------------------------------
// Kernel 2: init per-segment max / denominator (must run every call)
// ---------------------------------------------------------------------------
__global__ void seg_init(unsigned* __restrict__ m_ord, float* __restrict__ den, int Nn)
{
  int i = blockIdx.x * blockDim.x + threadIdx.x;
  if (i < Nn) { m_ord[i] = 0u; den[i] = 0.0f; }  // ordered-0 < every real float
}

// ---------------------------------------------------------------------------
// Kernel 3: per-edge raw logit + segment atomicMax
// ---------------------------------------------------------------------------
__global__ void edge_raw(const int* __restrict__ idx0, const int* __restrict__ idx1,
                         const float* __restrict__ from_pos, const float* __restrict__ to_pos,
                         const float* __restrict__ from_frame, const float* __restrict__ to_frame,
                         const float* __restrict__ Wattn,
                         const float* __restrict__ Anode, const float* __restrict__ Bnode,
                         float* __restrict__ raw, unsigned* __restrict__ m_ord, int Ee)
{
  int e = blockIdx.x * blockDim.x + threadIdx.x;
  if (e >= Ee) return;
  const int i = idx0[e];
  const int j = idx1[e];
  const float p0 = to_pos[j * 3 + 0] - from_pos[i * 3 + 0];
  const float p1 = to_pos[j * 3 + 1] - from_pos[i * 3 + 1];
  const float p2 = to_pos[j * 3 + 2] - from_pos[i * 3 + 2];

  float r = Anode[i] + Bnode[j];
#pragma unroll
  for (int d = 0; d < 3; ++d) {
    const float gf =  (p0 * from_frame[i * 9 + d] + p1 * from_frame[i * 9 + 3 + d] +
                       p2 * from_frame[i * 9 + 6 + d]);
    const float gt = -(p0 * to_frame[j * 9 + d] + p1 * to_frame[j * 9 + 3 + d] +
                       p2 * to_frame[j * 9 + 6 + d]);
    r += silu_f(gf) * Wattn[112 + d] + silu_f(gt) * Wattn[163 + d];
  }
  raw[e] = r;
  atomicMax(&m_ord[i], f32_ord(r));
}

// ---------------------------------------------------------------------------
// Kernel 4: exp(raw - segmax) + segment atomicAdd (ex overwrites raw)
// ---------------------------------------------------------------------------
__global__ void edge_expsum(const int* __restrict__ idx0,
                            const unsigned* __restrict__ m_ord,
                            float* __restrict__ raw, float* __restrict__ den, int Ee)
{
  int e = blockIdx.x * blockDim.x + threadIdx.x;
  if (e >= Ee) return;
  const int i = idx0[e];
  const float ex = expf(raw[e] - ord_f32(m_ord[i]));
  raw[e] = ex;
  atomicAdd(&den[i], ex);
}

// ---------------------------------------------------------------------------
// Kernel 5: normalize
// ---------------------------------------------------------------------------
__global__ void edge_norm(const int* __restrict__ idx0,
                          const float* __restrict__ raw, const float* __restrict__ den,
                          float* __restrict__ out, int Ee)
{
  int e = blockIdx.x * blockDim.x + threadIdx.x;
  if (e >= Ee) return;
  out[e] = raw[e] / den[idx0[e]];
}

// ---------------------------------------------------------------------------
extern "C" void kernel_launch(void* const* d_in, const int* in_sizes, int n_in,
                              void* d_out, int out_size, void* d_ws, size_t ws_size,
                              hipStream_t stream)
{
  const float* from_s     = (const float*)d_in[0];
  const float* from_v     = (const float*)d_in[1];
  const float* to_s       = (const float*)d_in[2];
  const float* to_v       = (const float*)d_in[3];
  const float* from_frame = (const float*)d_in[4];
  const float* to_frame   = (const float*)d_in[5];
  const float* from_pos   = (const float*)d_in[6];
  const float* to_pos     = (const float*)d_in[7];
  const float* Wfs        = (const float*)d_in[8];
  const float* Wts        = (const float*)d_in[9];
  const float* Wfv        = (const float*)d_in[10];
  const float* Wtv        = (const float*)d_in[11];
  const float* Wattn      = (const float*)d_in[12];
  const int*   inc        = (const int*)d_in[13];

  const int Nn = in_sizes[0] / 64;   // 50000
  const int Ee = in_sizes[13] / 2;   // 800000
  const int* idx0 = inc;
  const int* idx1 = inc + Ee;

  // workspace layout (floats): A[N] | B[N] | m_ord[N] | den[N] | raw[E]
  float*    ws    = (float*)d_ws;
  float*    Anode = ws;
  float*    Bnode = ws + Nn;
  unsigned* m_ord = (unsigned*)(ws + 2 * (size_t)Nn);
  float*    den   = ws + 3 * (size_t)Nn;
  float*    raw   = ws + 4 * (size_t)Nn;
  float*    out   = (float*)d_out;

  const int NT = (Nn + 15) / 16;

  node_precompute<<<(NT + WAVES_PER_BLK - 1) / WAVES_PER_BLK, 32 * WAVES_PER_BLK, 0, stream>>>(
      from_s, from_v, to_s, to_v, from_frame, to_frame,
      Wfs, Wts, Wfv, Wtv, Wattn, Anode, Bnode, Nn, NT);

  seg_init<<<(Nn + 255) / 256, 256, 0, stream>>>(m_ord, den, Nn);

  const int egrid = (Ee + 255) / 256;
  edge_raw<<<egrid, 256, 0, stream>>>(idx0, idx1, from_pos, to_pos, from_frame, to_frame,
                                      Wattn, Anode, Bnode, raw, m_ord, Ee);
  edge_expsum<<<egrid, 256, 0, stream>>>(idx0, m_ord, raw, den, Ee);
  edge_norm<<<egrid, 256, 0, stream>>>(idx0, raw, den, out, Ee);
}